// EmmaSAGELayer_15152644620657
// MI455X (gfx1250) — compile-verified
//
#include <hip/hip_runtime.h>

// EmmaSAGELayer for MI455X (gfx1250, wave32)
//   agg[i] = sum_{(i<-j)} x[j];  deg[i] = in-degree
//   out = inv_w * (agg @ W1^T) + x @ W2^T + bias,  W = [W1 | W2]
// Normalization folded into the GEMM epilogue so the LDS stage is a raw copy
// via GLOBAL_LOAD_ASYNC_TO_LDS_B128. GEMM via V_WMMA_F32_16X16X4_F32.

#define N_NODES 100000
#define C_IN    128
#define C_OUT   128
#define K_DIM   256   // 2*C_IN

typedef __attribute__((ext_vector_type(2))) float v2f;
typedef __attribute__((ext_vector_type(8))) float v8f;

#if __has_builtin(__builtin_amdgcn_global_load_async_to_lds_b128) && \
    __has_builtin(__builtin_amdgcn_s_wait_asynccnt)
#define USE_ASYNC_LDS 1
typedef int v4i_ls __attribute__((vector_size(16)));
typedef __attribute__((address_space(1))) v4i_ls* async_gptr;
typedef __attribute__((address_space(3))) v4i_ls* async_lptr;
#else
#define USE_ASYNC_LDS 0
#endif

// ---------------- zero workspace ----------------
__global__ void zero_ws(float* __restrict__ ws, int n) {
    int stride = gridDim.x * blockDim.x;
    for (int i = blockIdx.x * blockDim.x + threadIdx.x; i < n; i += stride)
        ws[i] = 0.0f;
}

// ---------------- edge scatter-add ----------------
// one wave32 per edge: 32 lanes x float4 = 128 channels
__global__ void __launch_bounds__(256) scatter_edges(
        const float* __restrict__ x,
        const int*   __restrict__ esrc,
        const int*   __restrict__ edst,
        float*       __restrict__ agg,
        float*       __restrict__ deg,
        int n_edges) {
    int edge = blockIdx.x * (blockDim.x >> 5) + (threadIdx.x >> 5);
    int lane = threadIdx.x & 31;
    if (edge >= n_edges) return;
    int s = esrc[edge];
    int d = edst[edge];
    const float4* xs = (const float4*)(x + (size_t)s * C_IN);
    float4 v = xs[lane];                       // channels lane*4 .. lane*4+3
    float* ad = agg + (size_t)d * C_IN + lane * 4;
    atomicAdd(ad + 0, v.x);
    atomicAdd(ad + 1, v.y);
    atomicAdd(ad + 2, v.z);
    atomicAdd(ad + 3, v.w);
    if (lane == 0) atomicAdd(deg + d, 1.0f);
}

// ---------------- concat-stage + GEMM (WMMA f32) ----------------
// block = 256 threads (8 waves). Block owns 16 nodes; wave w owns out-cols
// [16w, 16w+16). K=256 swept in steps of 4 -> 64 wmma_f32_16x16x4_f32 / wave,
// split into two accumulators (agg half / x half) so degree normalization can
// be applied per-row in the epilogue.
__global__ void __launch_bounds__(256) gemm_wmma(
        const float* __restrict__ x,
        const float* __restrict__ agg,
        const float* __restrict__ deg,
        const float* __restrict__ w,      // [C_OUT][K_DIM] row-major
        const float* __restrict__ bias,   // [C_OUT]
        float*       __restrict__ out) {  // [N][C_OUT]
    __shared__ float lds_h[16][K_DIM + 4];   // +4 pad: 4-bank row skew, conflict-free

    const int nodeBase = blockIdx.x * 16;
    const int tid = threadIdx.x;

    // ---- stage raw [agg | x] rows for 16 nodes into LDS (pure copy) ----
    // 16 rows x 64 chunks of 16B = 1024 transfers; 4 per thread, lane-coalesced.
#pragma unroll
    for (int it = 0; it < 4; ++it) {
        int idx = it * 256 + tid;              // 0..1023
        int row = idx >> 6;                    // node-in-tile
        int seg = idx & 63;                    // 16B chunk within 256-float row
        const float* src;
        int colf;                              // float column inside lds row
        if (seg < 32) { src = agg + (size_t)(nodeBase + row) * C_IN + seg * 4;
                        colf = seg * 4; }
        else          { src = x   + (size_t)(nodeBase + row) * C_IN + (seg - 32) * 4;
                        colf = C_IN + (seg - 32) * 4; }
#if USE_ASYNC_LDS
        __builtin_amdgcn_global_load_async_to_lds_b128(
            (async_gptr)(void*)(const_cast<float*>(src)),
            (async_lptr)(void*)&lds_h[row][colf],
            0, 0);
#else
        *(float4*)&lds_h[row][colf] = *(const float4*)src;
#endif
    }
#if USE_ASYNC_LDS
    __builtin_amdgcn_s_wait_asynccnt(0);
#endif
    __syncthreads();

    // ---- per-wave WMMA tiles ----
    const int wave    = tid >> 5;          // 0..7
    const int lane    = tid & 31;
    const int colBase = wave * 16;
    const int mn      = lane & 15;         // M for A-frag, N for B/C/D-frags
    const int kOff    = (lane >> 4) * 2;   // K sub-offset per ISA 16x4 f32 layout

    // B[k][n] = W^T[k][n] = W[colBase+n][k]
    const float* wrow = w + (size_t)(colBase + mn) * K_DIM;
    const float* arow = &lds_h[mn][0];

    v8f accA = {};   // agg half, K = 0..127
#pragma unroll 8
    for (int k0 = 0; k0 < C_IN; k0 += 4) {
        v2f a, b;
        a[0] = arow[k0 + kOff];
        a[1] = arow[k0 + kOff + 1];
        b[0] = wrow[k0 + kOff];
        b[1] = wrow[k0 + kOff + 1];
        accA = __builtin_amdgcn_wmma_f32_16x16x4_f32(
                false, a, false, b, (short)0, accA, false, false);
    }
    v8f accX = {};   // x half, K = 128..255
#pragma unroll 8
    for (int k0 = C_IN; k0 < K_DIM; k0 += 4) {
        v2f a, b;
        a[0] = arow[k0 + kOff];
        a[1] = arow[k0 + kOff + 1];
        b[0] = wrow[k0 + kOff];
        b[1] = wrow[k0 + kOff + 1];
        accX = __builtin_amdgcn_wmma_f32_16x16x4_f32(
                false, a, false, b, (short)0, accX, false, false);
    }

    // ---- epilogue: VGPR r holds M = r + 8*(lane>=16), N = lane&15 ----
    const float bv      = bias[colBase + mn];
    const int   rowHalf = (lane >> 4) * 8;
#pragma unroll
    for (int r = 0; r < 8; ++r) {
        int node  = nodeBase + rowHalf + r;
        float dg  = deg[node];
        float inw = dg > 0.0f ? 1.0f / dg : 0.0f;
        out[(size_t)node * C_OUT + colBase + mn] = inw * accA[r] + accX[r] + bv;
    }
}

// ---------------- launch ----------------
extern "C" void kernel_launch(void* const* d_in, const int* in_sizes, int n_in,
                              void* d_out, int out_size, void* d_ws, size_t ws_size,
                              hipStream_t stream) {
    const float* x    = (const float*)d_in[0];
    const int*   esrc = (const int*)  d_in[1];
    const int*   edst = (const int*)  d_in[2];
    const float* w    = (const float*)d_in[3];
    const float* bias = (const float*)d_in[4];
    float*       out  = (float*)d_out;

    float* agg = (float*)d_ws;                       // [N_NODES][C_IN]
    float* deg = agg + (size_t)N_NODES * C_IN;       // [N_NODES]
    int n_edges = in_sizes[1];

    int zn = N_NODES * C_IN + N_NODES;
    zero_ws<<<2048, 256, 0, stream>>>(agg, zn);

    int edgesPerBlock = 256 / 32;
    int sgrid = (n_edges + edgesPerBlock - 1) / edgesPerBlock;
    scatter_edges<<<sgrid, 256, 0, stream>>>(x, esrc, edst, agg, deg, n_edges);

    gemm_wmma<<<N_NODES / 16, 256, 0, stream>>>(x, agg, deg, w, bias, out);
}